// FPNBBoxHead_33706903339045
// MI455X (gfx1250) — compile-verified
//
#include <hip/hip_runtime.h>
#include <math.h>

// ---------------------------------------------------------------------------
// FPN BBox head for MI455X (gfx1250, wave32).
//   Stage 1: fused ROIAlign (level-select, bilinear, 7x7) + max-pool
//   Stage 2: h1 = relu(pooled @ W1 + b1)      (WMMA f32 16x16x4, double-buffered LDS)
//   Stage 3: h2 = relu(h1 @ W2 + b2)
//   Stage 4: heads -> sigmoid conf + box transform
// ---------------------------------------------------------------------------

typedef __attribute__((ext_vector_type(2))) float v2f;
typedef __attribute__((ext_vector_type(8))) float v8f;

#define NROI 2048   // B*P
#define CCH  256
#define HID  1024

// ----------------------------- ROIAlign + pool -----------------------------
__global__ __launch_bounds__(256) void roi_pool_kernel(
    const float* __restrict__ p0, const float* __restrict__ p1,
    const float* __restrict__ p2, const float* __restrict__ p3,
    const float* __restrict__ boxes, float* __restrict__ pooled)
{
    const int bp = blockIdx.x;        // ROI index 0..2047
    const int b  = bp >> 9;           // P = 512
    const int c  = threadIdx.x;       // channel 0..255

    const float y0 = boxes[bp*4+0], x0 = boxes[bp*4+1];
    const float y1 = boxes[bp*4+2], x1 = boxes[bp*4+3];
    const float bh = y1 - y0, bw = x1 - x0;
    const float area = fmaxf(bh*bw, 1e-8f);
    int lvl = (int)floorf(log2f(sqrtf(area) / 224.f) + 4.f);
    lvl = lvl < 1 ? 1 : (lvl > 4 ? 4 : lvl);

    const float* feat; int h;
    switch (lvl) {
        case 1:  feat = p0; h = 256; break;
        case 2:  feat = p1; h = 128; break;
        case 3:  feat = p2; h = 64;  break;
        default: feat = p3; h = 32;  break;
    }
    const int w = h;
    const float inv = 1.f / (float)(1 << lvl);
    const float ys = y0*inv, xs = x0*inv, bhs = bh*inv, bws = bw*inv;

    int   yi0[7], yi1[7], xi0[7], xi1[7];
    float ly[7], lx[7];
#pragma unroll
    for (int i = 0; i < 7; ++i) {
        const float t  = ((float)i + 0.5f) / 7.f;
        const float gy = fminf(fmaxf(ys + t*bhs, 0.f), (float)(h-1));
        const float gx = fminf(fmaxf(xs + t*bws, 0.f), (float)(w-1));
        const int yy = (int)floorf(gy), xx = (int)floorf(gx);
        yi0[i] = yy; yi1[i] = (yy+1 < h-1) ? yy+1 : h-1;
        xi0[i] = xx; xi1[i] = (xx+1 < w-1) ? xx+1 : w-1;
        ly[i] = gy - (float)yy; lx[i] = gx - (float)xx;
    }

    const float* base = feat + ((size_t)b * h * w) * CCH + c;
    float mv = -3.4e38f;
#pragma unroll
    for (int iy = 0; iy < 7; ++iy) {
        const float* rA = base + (size_t)yi0[iy] * w * CCH;
        const float* rB = base + (size_t)yi1[iy] * w * CCH;
        const float wy = ly[iy];
#pragma unroll
        for (int ix = 0; ix < 7; ++ix) {
            const float wx  = lx[ix];
            const float f00 = rA[(size_t)xi0[ix]*CCH];
            const float f01 = rA[(size_t)xi1[ix]*CCH];
            const float f10 = rB[(size_t)xi0[ix]*CCH];
            const float f11 = rB[(size_t)xi1[ix]*CCH];
            const float top = f00 + wx*(f01 - f00);
            const float bot = f10 + wx*(f11 - f10);
            mv = fmaxf(mv, top + wy*(bot - top));
        }
    }
    pooled[(size_t)bp*CCH + c] = mv;
}

// ------------------------------ WMMA f32 GEMM -------------------------------
// D[M,N] = act(A[M,K] @ Bm[K,N] + bias).
// Block tile 64(M) x 128(N), K tile 32, 8 wave32s; each wave computes 16x64.
// A staged row-major (K contiguous -> ds_load_2addr_b64 fragments);
// B staged TRANSPOSED in LDS (BsT[n][k], stride 34) so each B fragment
// {k,k+1} is a single aligned ds_load_b64 into an even VGPR pair — no
// packing moves ahead of the WMMA burst. Stride 34 makes the 32-lane b64
// fragment read touch all 64 banks exactly once (conflict-free).
#define BM 64
#define BN 128
#define BK 32
#define APAD 4   // A row stride 36 floats: 16B-aligned rows, conflict-free frag reads
#define BTPAD 2  // BsT row stride 34 floats: 8B-aligned, all-64-bank b64 reads

__global__ __launch_bounds__(256) void gemm_bias_act_kernel(
    const float* __restrict__ A, const float* __restrict__ Bm,
    const float* __restrict__ bias, float* __restrict__ D,
    int M, int N, int K, int relu)
{
    __shared__ __align__(16) float As [2][BM][BK + APAD];
    __shared__ __align__(16) float BsT[2][BN][BK + BTPAD];   // [n][k]

    const int tid  = threadIdx.x;
    const int lane = tid & 31;
    const int wid  = tid >> 5;
    const int half = lane >> 4;        // selects K pair / M high-half per ISA layout
    const int ml   = lane & 15;
    const int msub = wid >> 1;         // 0..3 : 16-row M subtile
    const int ncolw = (wid & 1) * 64;  // wave's 64-col window within block tile
    const int m0 = blockIdx.y * BM;
    const int n0 = blockIdx.x * BN;
    const int arow = msub * 16 + ml;

    // Staging indices (float4-coalesced global loads)
    const int a_r  = tid >> 3;          // 0..31, +32 stride
    const int a_c4 = (tid & 7) * 4;     // 0..28
    const int b_r  = tid >> 5;          // 0..7,  +8 stride  (k index)
    const int b_c4 = (tid & 31) * 4;    // 0..124            (n index)

    v8f acc0 = {}, acc1 = {}, acc2 = {}, acc3 = {};
    float4 ar[2], br[4];

    const int nt = K / BK;

    // Prologue: fetch tile 0 and stage into LDS buffer 0.
#pragma unroll
    for (int i = 0; i < 2; ++i)
        ar[i] = *(const float4*)&A[(size_t)(m0 + a_r + 32*i) * K + a_c4];
#pragma unroll
    for (int i = 0; i < 4; ++i)
        br[i] = *(const float4*)&Bm[(size_t)(b_r + 8*i) * N + n0 + b_c4];
#pragma unroll
    for (int i = 0; i < 2; ++i)
        *(float4*)&As[0][a_r + 32*i][a_c4] = ar[i];
#pragma unroll
    for (int i = 0; i < 4; ++i) {
        const int r = b_r + 8*i;
        BsT[0][b_c4 + 0][r] = br[i].x;
        BsT[0][b_c4 + 1][r] = br[i].y;
        BsT[0][b_c4 + 2][r] = br[i].z;
        BsT[0][b_c4 + 3][r] = br[i].w;
    }

    for (int t = 0; t < nt; ++t) {
        const int buf = t & 1;
        const int k0n = (t + 1) * BK;

        // Issue global loads for the next tile (overlap with compute).
        if (t + 1 < nt) {
#pragma unroll
            for (int i = 0; i < 2; ++i)
                ar[i] = *(const float4*)&A[(size_t)(m0 + a_r + 32*i) * K + k0n + a_c4];
#pragma unroll
            for (int i = 0; i < 4; ++i)
                br[i] = *(const float4*)&Bm[(size_t)(k0n + b_r + 8*i) * N + n0 + b_c4];
        }

        __syncthreads();   // LDS buffer buf ready; prior reads of the other LDS buffer retired

        // Preload all fragments of this K tile into distinct registers.
        v2f af[8];
#pragma unroll
        for (int s = 0; s < 8; ++s) {
            const int k = 4*s + 2*half;
            af[s] = *(const v2f*)&As[buf][arow][k];
        }
        v2f bf[4][8];
#pragma unroll
        for (int s = 0; s < 8; ++s) {
            const int k = 4*s + 2*half;
#pragma unroll
            for (int j = 0; j < 4; ++j)
                bf[j][s] = *(const v2f*)&BsT[buf][ncolw + 16*j + ml][k];
        }

        // WMMA burst: 32 matrix ops, 4 independent accumulation chains.
#pragma unroll
        for (int s = 0; s < 8; ++s) {
            acc0 = __builtin_amdgcn_wmma_f32_16x16x4_f32(
                false, af[s], false, bf[0][s], (short)0, acc0, false, false);
            acc1 = __builtin_amdgcn_wmma_f32_16x16x4_f32(
                false, af[s], false, bf[1][s], (short)0, acc1, false, false);
            acc2 = __builtin_amdgcn_wmma_f32_16x16x4_f32(
                false, af[s], false, bf[2][s], (short)0, acc2, false, false);
            acc3 = __builtin_amdgcn_wmma_f32_16x16x4_f32(
                false, af[s], false, bf[3][s], (short)0, acc3, false, false);
        }

        // Stage next tile into the other LDS buffer (in the compute shadow).
        if (t + 1 < nt) {
            const int nb = buf ^ 1;
#pragma unroll
            for (int i = 0; i < 2; ++i)
                *(float4*)&As[nb][a_r + 32*i][a_c4] = ar[i];
#pragma unroll
            for (int i = 0; i < 4; ++i) {
                const int r = b_r + 8*i;
                BsT[nb][b_c4 + 0][r] = br[i].x;
                BsT[nb][b_c4 + 1][r] = br[i].y;
                BsT[nb][b_c4 + 2][r] = br[i].z;
                BsT[nb][b_c4 + 3][r] = br[i].w;
            }
        }
    }

    // Epilogue: D layout -> VGPR i holds M = 16*msub + 8*half + i, N = ml.
#pragma unroll
    for (int i = 0; i < 8; ++i) {
        const int row = m0 + msub*16 + half*8 + i;
#pragma unroll
        for (int j = 0; j < 4; ++j) {
            const int col = n0 + ncolw + 16*j + ml;
            float v = (j == 0 ? acc0[i] : j == 1 ? acc1[i] : j == 2 ? acc2[i] : acc3[i])
                      + bias[col];
            if (relu) v = fmaxf(v, 0.f);
            D[(size_t)row * N + col] = v;
        }
    }
}

// ------------------------------- Output heads -------------------------------
__device__ __forceinline__ float sigmoidf_(float x) { return 1.f / (1.f + expf(-x)); }

__global__ __launch_bounds__(256) void head_kernel(
    const float* __restrict__ H,  const float* __restrict__ Wc,
    const float* __restrict__ bc, const float* __restrict__ Wb,
    const float* __restrict__ bb, float* __restrict__ out, int nrows)
{
    const int row  = blockIdx.x * (blockDim.x >> 5) + (threadIdx.x >> 5);
    const int lane = threadIdx.x & 31;
    if (row >= nrows) return;

    const float* h = H + (size_t)row * HID;
    float sc = 0.f, s0 = 0.f, s1 = 0.f, s2 = 0.f, s3 = 0.f;
    for (int j = lane; j < HID; j += 32) {
        const float v = h[j];
        sc += v * Wc[j];
        s0 += v * Wb[j*4+0];
        s1 += v * Wb[j*4+1];
        s2 += v * Wb[j*4+2];
        s3 += v * Wb[j*4+3];
    }
#pragma unroll
    for (int off = 16; off > 0; off >>= 1) {
        sc += __shfl_down(sc, off, 32);
        s0 += __shfl_down(s0, off, 32);
        s1 += __shfl_down(s1, off, 32);
        s2 += __shfl_down(s2, off, 32);
        s3 += __shfl_down(s3, off, 32);
    }
    if (lane == 0) {
        out[row] = sigmoidf_(sc + bc[0]);
        const float ry = sigmoidf_(s0 + bb[0]);
        const float rx = sigmoidf_(s1 + bb[1]);
        const float hy = sigmoidf_(s2 + bb[2]) * 0.5f;
        const float hx = sigmoidf_(s3 + bb[3]) * 0.5f;
        float* bx = out + nrows + (size_t)row * 4;
        bx[0] = fminf(fmaxf(ry - hy, 0.f), 1.f);
        bx[1] = fminf(fmaxf(rx - hx, 0.f), 1.f);
        bx[2] = fminf(fmaxf(ry + hy, 0.f), 1.f);
        bx[3] = fminf(fmaxf(rx + hx, 0.f), 1.f);
    }
}

// --------------------------------- Launch -----------------------------------
extern "C" void kernel_launch(void* const* d_in, const int* in_sizes, int n_in,
                              void* d_out, int out_size, void* d_ws, size_t ws_size,
                              hipStream_t stream)
{
    const float* p0    = (const float*)d_in[0];
    const float* p1    = (const float*)d_in[1];
    const float* p2    = (const float*)d_in[2];
    const float* p3    = (const float*)d_in[3];
    const float* boxes = (const float*)d_in[4];
    const float* W1    = (const float*)d_in[5];
    const float* b1    = (const float*)d_in[6];
    const float* W2    = (const float*)d_in[7];
    const float* b2    = (const float*)d_in[8];
    const float* Wc    = (const float*)d_in[9];
    const float* bc    = (const float*)d_in[10];
    const float* Wb    = (const float*)d_in[11];
    const float* bb    = (const float*)d_in[12];

    float* pooled = (float*)d_ws;                       // 2048*256  = 2 MB
    float* h1     = pooled + (size_t)NROI * CCH;        // 2048*1024 = 8 MB
    float* h2     = h1     + (size_t)NROI * HID;        // 2048*1024 = 8 MB
    float* out    = (float*)d_out;

    roi_pool_kernel<<<NROI, 256, 0, stream>>>(p0, p1, p2, p3, boxes, pooled);

    dim3 g(HID / BN, NROI / BM);   // (8, 32)
    gemm_bias_act_kernel<<<g, 256, 0, stream>>>(pooled, W1, b1, h1, NROI, HID, CCH, 1);
    gemm_bias_act_kernel<<<g, 256, 0, stream>>>(h1,     W2, b2, h2, NROI, HID, HID, 1);

    head_kernel<<<NROI / 8, 256, 0, stream>>>(h2, Wc, bc, Wb, bb, out, NROI);
}